// TensorizedLinear_module_30365418783362
// MI455X (gfx1250) — compile-verified
//
#include <hip/hip_runtime.h>

// TT tensorized linear for MI455X (gfx1250, wave32).
// Memory-bound: 256 MiB stream / 23.3 TB/s ~= 11.5 us floor. All math fp32 via
// V_WMMA_F32_16X16X4_F32 (rank 20 padded to 32). A and C are pre-swizzled into
// the WMMA B-fragment order so every LDS fragment fetch is one aligned
// ds_load_b64 (conflict-free across all 64 banks, no VALU repacking).

typedef float v2f __attribute__((ext_vector_type(2)));
typedef float v4f __attribute__((ext_vector_type(4)));
typedef float v8f __attribute__((ext_vector_type(8)));

#define TTR 20      // true TT rank
#define KF  1024    // input features
#define NF  1024    // output features
#define RP  32      // padded rank

// Swizzled B-fragment layouts (float index):
//   Apack[kk*128  + n*4 + half*2 + j] = A[4*kk + 2*half + j][n]   kk<256, n<32
//   Cpack[kk*4096 + n*4 + half*2 + j] = C[4*kk + 2*half + j][n]   kk<8,  n<1024

// ---------------------------------------------------------------------------
// Kernel 1: merge TT cores directly into swizzled layouts.
//   A[k][r] = sum_c f0[0, k>>5, c] * f1[c, k&31, r]   (r>=20 -> 0)
//   C[d][n] = sum_e f2[d, n>>5, e] * f3[e, n&31, 0]   (d>=20 -> 0)
// ---------------------------------------------------------------------------
__global__ __launch_bounds__(256) void tt_precompute(
    const float* __restrict__ f0, const float* __restrict__ f1,
    const float* __restrict__ f2, const float* __restrict__ f3,
    float* __restrict__ Apack, float* __restrict__ Cpack)
{
  const int i = blockIdx.x * 256 + threadIdx.x;   // 0..32767
  {
    const int kk = i >> 7, rem = i & 127;
    const int n = rem >> 2, half = (rem >> 1) & 1, j = rem & 1;
    const int k = 4 * kk + 2 * half + j;          // input-feature row
    const int m0 = k >> 5, m1 = k & 31;
    float acc = 0.0f;
    if (n < TTR) {
#pragma unroll
      for (int c = 0; c < TTR; ++c)
        acc = fmaf(f0[m0 * TTR + c], f1[(c * 32 + m1) * TTR + n], acc);
    }
    Apack[i] = acc;
  }
  {
    const int kk = i >> 12, rem = i & 4095;
    const int n = rem >> 2, half = (rem >> 1) & 1, j = rem & 1;
    const int d = 4 * kk + 2 * half + j;          // rank row
    const int p = n >> 5, q = n & 31;
    float acc = 0.0f;
    if (d < TTR) {
#pragma unroll
      for (int e = 0; e < TTR; ++e)
        acc = fmaf(f2[(d * 32 + p) * TTR + e], f3[e * 32 + q], acc);
    }
    Cpack[i] = acc;
  }
}

// ---------------------------------------------------------------------------
// Kernel 2: core[t][r] = sum_k x[t][k] * A[k][r]
// 256 blocks x 256 threads (8 waves); wave owns 16 token rows.
// 256 k-steps (K=4) x 2 rank tiles -> 512 v_wmma_f32_16x16x4_f32 per wave.
// ---------------------------------------------------------------------------
__global__ __launch_bounds__(256) void tt_stage2(
    const float* __restrict__ x, const float* __restrict__ Apack,
    float* __restrict__ core)
{
  __shared__ float A_s[KF * RP];                  // 131072 B, swizzled

  const int tid = threadIdx.x;
  {
    v4f* __restrict__ dst = (v4f*)A_s;
    const v4f* __restrict__ src = (const v4f*)Apack;
    for (int idx = tid; idx < KF * RP / 4; idx += 256) dst[idx] = src[idx];
  }
  __syncthreads();

  const int wid  = tid >> 5, lane = tid & 31;
  const int half = lane >> 4, lrow = lane & 15;
  const long row0 = (long)blockIdx.x * 128 + (long)wid * 16;

  // A-operand: lanes 0-15 -> row M=lane, K={0,1}; lanes 16-31 -> K={2,3}.
  const float* xrow = x + (row0 + lrow) * KF + half * 2;
  const int bofs = lrow * 4 + half * 2;           // lane offset into Apack block

  v8f acc0 = {};   // rank cols 0..15
  v8f acc1 = {};   // rank cols 16..31 (>=20 stay zero: A cols are zero)
#pragma unroll 8
  for (int kk = 0; kk < KF / 4; ++kk) {
    const v2f a  = __builtin_nontemporal_load((const v2f*)(xrow + kk * 4));
    const v2f b0 = *(const v2f*)(&A_s[kk * 128 + bofs]);       // n = lrow
    const v2f b1 = *(const v2f*)(&A_s[kk * 128 + bofs + 64]);  // n = lrow+16
    acc0 = __builtin_amdgcn_wmma_f32_16x16x4_f32(false, a, false, b0,
                                                 (short)0, acc0, false, false);
    acc1 = __builtin_amdgcn_wmma_f32_16x16x4_f32(false, a, false, b1,
                                                 (short)0, acc1, false, false);
  }

  // D layout: VGPR v -> (M=v, N=lane) lanes 0-15; (M=v+8, N=lane-16) lanes 16-31.
  float* cb = core + row0 * RP + lrow;
#pragma unroll
  for (int v = 0; v < 8; ++v) {
    const int m = v + half * 8;
    cb[m * RP]      = acc0[v];
    cb[m * RP + 16] = acc1[v];
  }
}

// ---------------------------------------------------------------------------
// Kernel 3: out[t][n] = sum_d core[t][d] * C[d][n] + bias[n]
// Core A-fragments held in registers; 64 n-tiles x 8 k-steps -> 512 WMMAs/wave.
// ---------------------------------------------------------------------------
__global__ __launch_bounds__(256) void tt_stage4(
    const float* __restrict__ core, const float* __restrict__ Cpack,
    const float* __restrict__ bias, float* __restrict__ out)
{
  __shared__ float C_s[RP * NF];                  // 131072 B, swizzled

  const int tid = threadIdx.x;
  {
    v4f* __restrict__ dst = (v4f*)C_s;
    const v4f* __restrict__ src = (const v4f*)Cpack;
    for (int idx = tid; idx < RP * NF / 4; idx += 256) dst[idx] = src[idx];
  }
  __syncthreads();

  const int wid  = tid >> 5, lane = tid & 31;
  const int half = lane >> 4, lrow = lane & 15;
  const long row0 = (long)blockIdx.x * 128 + (long)wid * 16;

  // Hoist this wave's core A-fragments (16 rows x 32 k) into registers.
  v2f af[8];
  const float* crow = core + (row0 + lrow) * RP + half * 2;
#pragma unroll
  for (int kk = 0; kk < 8; ++kk) af[kk] = *(const v2f*)(crow + kk * 4);

  const int bofs = lrow * 4 + half * 2;
  float* ob = out + row0 * NF + lrow;
  for (int nt = 0; nt < NF / 16; ++nt) {
    const int n0 = nt * 16;
    const float b = bias[n0 + lrow];
    v8f o = { b, b, b, b, b, b, b, b };   // fold bias into accumulator init
#pragma unroll
    for (int kk = 0; kk < 8; ++kk) {
      const v2f bv = *(const v2f*)(&C_s[kk * 4096 + n0 * 4 + bofs]);
      o = __builtin_amdgcn_wmma_f32_16x16x4_f32(false, af[kk], false, bv,
                                                (short)0, o, false, false);
    }
#pragma unroll
    for (int v = 0; v < 8; ++v) {
      const int m = v + half * 8;
      __builtin_nontemporal_store(o[v], ob + m * NF + n0);
    }
  }
}

// ---------------------------------------------------------------------------
extern "C" void kernel_launch(void* const* d_in, const int* in_sizes, int n_in,
                              void* d_out, int out_size, void* d_ws, size_t ws_size,
                              hipStream_t stream)
{
  (void)in_sizes; (void)n_in; (void)out_size; (void)ws_size;
  const float* x    = (const float*)d_in[0];
  const float* f0   = (const float*)d_in[1];
  const float* f1   = (const float*)d_in[2];
  const float* f2   = (const float*)d_in[3];
  const float* f3   = (const float*)d_in[4];
  const float* bias = (const float*)d_in[5];

  // Workspace layout (floats): Apack 32768 | Cpack 32768 | core 32768*32
  float* Apack = (float*)d_ws;
  float* Cpack = Apack + KF * RP;
  float* core  = Cpack + RP * NF;

  tt_precompute<<<128, 256, 0, stream>>>(f0, f1, f2, f3, Apack, Cpack);
  tt_stage2<<<256, 256, 0, stream>>>(x, Apack, core);
  tt_stage4<<<256, 256, 0, stream>>>(core, Cpack, bias, (float*)d_out);
}